// PoissonIntervalEncoder_69071664054971
// MI455X (gfx1250) — compile-verified
//
#include <hip/hip_runtime.h>
#include <hip/hip_bf16.h>
#include <stdint.h>

// Poisson interval spike encoder for MI455X (gfx1250, wave32).
//
// Output [500, 65536] f32 = 131 MB of pure writes -> store-bandwidth bound
// (floor ~5.6us @ 23.3 TB/s). Each 256-thread block owns 256 columns,
// simulates the renewal process lazily (next-spike-time register, one compare
// per output row), expands 16 rows x 256 cols into a double-buffered LDS
// tile, and hands each tile to the Tensor Data Mover: ONE
// tensor_store_from_lds per tile per block (2D D# descriptor, dim0 stride =
// 65536 floats) drains LDS to HBM while all 8 waves keep sampling. Buffer
// reuse is fenced with s_wait_tensorcnt. The tiny input tile (1KB/block) is
// staged via the async engine (global_load_async_to_lds_b32 + asynccnt).
// Every output element is written exactly once; no zero-fill pass, no
// scattered read-modify-write.

#define STEPS   500
#define BLOCK   256
#define TILE_T  16                       // rows per LDS tile (16KB per buffer)

typedef unsigned int u32x4 __attribute__((ext_vector_type(4)));
typedef int          i32x4 __attribute__((ext_vector_type(4)));
typedef int          i32x8 __attribute__((ext_vector_type(8)));

// --- tiny counter-free hash RNG (no bit-exact match to JAX threefry; we
// --- only need a statistically sound Poisson process) ----------------------
__device__ __forceinline__ uint32_t rng_next(uint32_t& s) {
  s = s * 1664525u + 1013904223u;
  uint32_t x = s;
  x ^= x >> 17; x *= 0xed5ad4bbu;
  x ^= x >> 11; x *= 0xac4c1b51u;
  x ^= x >> 15;
  return x;
}
__device__ __forceinline__ float uniform01(uint32_t& s) {
  // (0,1), never 0 so log() is safe
  return ((float)(rng_next(s) >> 8) + 0.5f) * 5.9604644775390625e-08f; // *2^-24
}

// One inter-spike interval ~ Poisson(lam), clamped to >= 1 (the reference
// replaces zero intervals with 1 for active elements).
__device__ __forceinline__ float draw_interval(float lam, float expNegLam,
                                               float sd, bool small_lam,
                                               uint32_t& st) {
  float k;
  if (small_lam) {
    // sequential-search inversion; lam <= 60 so exp(-lam) is a normal float
    float u = uniform01(st);
    float p = expNegLam, F = expNegLam;
    k = 0.0f;
    while (u > F && k < 320.0f) {
      k += 1.0f;
      p *= __fdividef(lam, k);
      F += p;
    }
  } else {
    // normal approximation (lam > 60): k = round(lam + sqrt(lam)*z)
    float u1 = uniform01(st), u2 = uniform01(st);
    float z = sqrtf(fmaxf(-2.0f * __logf(u1), 0.0f)) * __cosf(6.28318530718f * u2);
    k = floorf(lam + sd * z + 0.5f);
  }
  return fmaxf(k, 1.0f);
}

// Issue a 2D TDM store: LDS tile (rows x 256 f32, contiguous) -> global at
// gaddr with row stride of `stride` elements. Wave-0 only; TENSORcnt-tracked.
__device__ __forceinline__ void tdm_store_tile(uint32_t lds_byte_addr,
                                               uint64_t gaddr, int rows,
                                               int stride) {
  u32x4 g0;
  g0.x = 1u;                                           // count=1 valid user D#
  g0.y = lds_byte_addr;                                // lds_addr
  g0.z = (uint32_t)gaddr;                              // global_addr[31:0]
  g0.w = (uint32_t)((gaddr >> 32) & 0x01FFFFFFu)       // global_addr[56:32]
       | (2u << 30);                                   // type=2 ("image")
  i32x8 g1;
  g1[0] = (int)(2u << 16);                             // data_size=2 -> 4 bytes
  g1[1] = (int)(256u << 16);                           // tensor_dim0 = 256 (lo16)
  g1[2] = (int)(((uint32_t)rows & 0xFFFFu) << 16);     // dim0 hi=0 | tensor_dim1 lo16
  g1[3] = (int)(256u << 16);                           // tensor_dim1 hi=0 | tile_dim0=256
  g1[4] = (int)((uint32_t)rows & 0xFFFFu);             // tile_dim1=rows, tile_dim2=0
  g1[5] = stride;                                      // tensor_dim0_stride lo32
  g1[6] = 0;                                           // stride hi16 | dim1_stride lo16
  g1[7] = 0;
  i32x4 gz = {0, 0, 0, 0};                             // unused higher-dim groups
#if defined(__clang_major__) && (__clang_major__ >= 23)
  i32x8 gz8 = {0, 0, 0, 0, 0, 0, 0, 0};
  __builtin_amdgcn_tensor_store_from_lds(g0, g1, gz, gz, gz8, 0);
#else
  __builtin_amdgcn_tensor_store_from_lds(g0, g1, gz, gz, 0);
#endif
}

__global__ __launch_bounds__(BLOCK)
void poisson_interval_encoder_kernel(const float* __restrict__ in,
                                     float* __restrict__ out,
                                     int n_elems) {
  const int tid  = threadIdx.x;
  const int col0 = blockIdx.x * BLOCK;
  const int elem = col0 + tid;
  const bool wave0 = tid < 32;

  __shared__ __align__(16) float tile[2][TILE_T * BLOCK];  // 2 x 16 KB
  __shared__ __align__(16) float lds_in[BLOCK];            // 1 KB input stage
  // flat LDS aperture: low 32 bits of the generic address == LDS byte offset
  const uint32_t ldsbase[2] = {
    (uint32_t)(uintptr_t)&tile[0][0],
    (uint32_t)(uintptr_t)&tile[1][0]
  };
  const uint32_t lds_in_base = (uint32_t)(uintptr_t)&lds_in[0];

  // ---- stage input through the async engine (gather -> LDS, no VGPR hop) --
  {
    const uint32_t li = lds_in_base + (uint32_t)tid * 4u;
    const float*  gsrc = in + ((elem < n_elems) ? elem : 0);
    asm volatile("global_load_async_to_lds_b32 %0, %1, off"
                 :: "v"(li), "v"(gsrc)
                 : "memory");
    asm volatile("s_wait_asynccnt 0x0" ::: "memory");
  }
  __syncthreads();

  // ---- per-element renewal process state ---------------------------------
  float x    = (elem < n_elems) ? lds_in[tid] : 0.0f;
  float rate = 0.128f * x;                       // FREQ*DT/1000
  const bool  active    = rate > 0.0f;
  const float lam       = active ? fminf(1.0f / fmaxf(rate, 1e-12f), 1e9f) : 0.0f;
  const bool  small_lam = lam <= 60.0f;
  const float expNegLam = small_lam ? __expf(-lam) : 0.0f;
  const float sd        = small_lam ? 0.0f : sqrtf(lam);

  uint32_t st = (uint32_t)elem * 0x9E3779B9u + 0x0000002Au; // seed 42 mix
  st ^= st >> 16; st *= 0x85ebca6bu; st ^= st >> 13;

  // next spike time (integer-valued float); strictly increasing, >= 1
  float nxt = active ? draw_interval(lam, expNegLam, sd, small_lam, st) : 1e30f;

  const bool full_block = (col0 + BLOCK) <= n_elems;

  // ---- double-buffered LDS tiles, drained by the Tensor Data Mover -------
  const int nchunks = (STEPS + TILE_T - 1) / TILE_T;   // 32 (31 full + 4-row tail)
  for (int c = 0; c < nchunks; ++c) {
    const int buf  = c & 1;
    const int rows = min(TILE_T, STEPS - c * TILE_T);

    // before refilling this buffer, wave0's TDM store of 2 chunks ago must
    // have drained it; everyone else learns via the barrier
    if (c >= 2 && wave0)
      __builtin_amdgcn_s_wait_tensorcnt((short)1);
    __syncthreads();

    // fill: one column per thread (stride-4B across lanes -> conflict-free)
    float* tp = &tile[buf][0];
    #pragma unroll
    for (int r = 0; r < TILE_T; ++r) {
      if (r >= rows) break;
      const float tt = (float)(c * TILE_T + r + 1);    // reference time value
      float v = 0.0f;
      if (nxt == tt) {                                 // exact int compare in f32
        v = 1.0f;
        nxt += draw_interval(lam, expNegLam, sd, small_lam, st);
      }
      tp[r * BLOCK + tid] = v;
    }
    __syncthreads();

    if (full_block) {
      if (wave0) {
        // one TDM instruction moves the whole rows x 256 f32 tile to HBM
        uint64_t ga = (uint64_t)(uintptr_t)(out
                        + (size_t)(c * TILE_T) * (size_t)n_elems + col0);
        tdm_store_tile(ldsbase[buf], ga, rows, n_elems);
      }
    } else {
      // ragged tail block (n_elems not a multiple of 256): guarded scalars
      if (elem < n_elems) {
        for (int r = 0; r < rows; ++r)
          out[(size_t)(c * TILE_T + r) * (size_t)n_elems + elem] =
              tp[r * BLOCK + tid];
      }
    }
  }
  // drain all outstanding tensor stores before wave exit
  if (wave0)
    __builtin_amdgcn_s_wait_tensorcnt((short)0);
}

extern "C" void kernel_launch(void* const* d_in, const int* in_sizes, int n_in,
                              void* d_out, int out_size, void* d_ws, size_t ws_size,
                              hipStream_t stream) {
  const float* in  = (const float*)d_in[0];
  float*       out = (float*)d_out;
  const int n      = in_sizes[0];                 // 16*4096 = 65536 elements
  const int blocks = (n + BLOCK - 1) / BLOCK;     // 256 blocks x 256 threads (8 waves)
  poisson_interval_encoder_kernel<<<blocks, BLOCK, 0, stream>>>(in, out, n);
}